// GCN_23407571763562
// MI455X (gfx1250) — compile-verified
//
#include <hip/hip_runtime.h>

typedef float v2f __attribute__((ext_vector_type(2)));
typedef float v4f __attribute__((ext_vector_type(4)));
typedef float v8f __attribute__((ext_vector_type(8)));

// ---------------------------------------------------------------- degrees ---
__global__ __launch_bounds__(256) void gcn_deg_init(float* __restrict__ deg, int n) {
    int i = blockIdx.x * blockDim.x + threadIdx.x;
    if (i < n) deg[i] = 1.0f;                 // self-loop
}

__global__ __launch_bounds__(256) void gcn_deg_count(float* __restrict__ deg,
                                                     const int* __restrict__ dst, int e) {
    int i = blockIdx.x * blockDim.x + threadIdx.x;
    if (i < e) atomicAdd(&deg[dst[i]], 1.0f);
}

__global__ __launch_bounds__(256) void gcn_rsqrt(float* __restrict__ dis,
                                                 const float* __restrict__ deg, int n) {
    int i = blockIdx.x * blockDim.x + threadIdx.x;
    if (i < n) dis[i] = rsqrtf(deg[i]);
}

// ------------------------------------------------------- WMMA fp32 GEMM ------
// C[M,Nc] = op(A[M,KK] @ B[KK,Nc] + bias).
// Block = 256 threads = 8 waves. Weights B staged once into LDS, transposed
// with padded stride (KK+2) so a B fragment is a single aligned ds_load_b64.
// Each wave computes 16 rows x full Nc: TN accumulators (v8f each), the A
// fragment is loaded once per k-step and reused across all TN WMMAs.
template <int KK, int TN, bool RELU, bool BIAS>
__global__ __launch_bounds__(256) void gcn_gemm_wmma(
        const float* __restrict__ A, const float* __restrict__ B,
        const float* __restrict__ bias, float* __restrict__ C, int M) {
    constexpr int Nc  = TN * 16;
    constexpr int LDB = KK + 2;                 // pad: banks stride 2, 8B aligned
    __shared__ float Blds[Nc * LDB];

    // Stage B transposed: Blds[c*LDB + k] = B[k*Nc + c]
    // Global reads coalesced; LDS writes hit 32 distinct banks per half-wave.
    for (int i = threadIdx.x; i < KK * Nc; i += 256) {
        int k = i / Nc;
        int c = i - k * Nc;
        Blds[c * LDB + k] = B[i];
    }
    __syncthreads();

    int tilesM = M >> 4;
    int tm     = blockIdx.x * 8 + (threadIdx.x >> 5);
    int lane   = threadIdx.x & 31;
    int half   = lane >> 4;                     // K-pair select (ISA 7.12.2)
    int lr     = lane & 15;                     // row (A) / col (B,C)

    if (tm < tilesM) {                          // wave-uniform guard
        const float* Arow = A + (size_t)(tm * 16 + lr) * KK;
        __builtin_prefetch(Arow, 0, 0);         // global_prefetch_b8
        __builtin_prefetch(Arow + 64, 0, 0);

        v8f acc[TN] = {};
        for (int k = 0; k < KK; k += 4) {
            // A frag: lanes 0-15 -> K=k,k+1 ; lanes 16-31 -> K=k+2,k+3
            v2f a = *(const v2f*)(Arow + k + 2 * half);
#pragma unroll
            for (int t = 0; t < TN; ++t) {
                v2f b = *(const v2f*)(&Blds[(t * 16 + lr) * LDB + k + 2 * half]);
                acc[t] = __builtin_amdgcn_wmma_f32_16x16x4_f32(
                             false, a, false, b, (short)0, acc[t], false, false);
            }
        }

#pragma unroll
        for (int t = 0; t < TN; ++t) {
            int col  = t * 16 + lr;
            float bv = BIAS ? bias[col] : 0.0f;
#pragma unroll
            for (int r = 0; r < 8; ++r) {
                int row = tm * 16 + r + 8 * half;   // C/D layout (ISA 7.12.2)
                float v = acc[t][r] + bv;
                if (RELU) v = fmaxf(v, 0.0f);
                C[(size_t)row * Nc + col] = v;
            }
        }
    }
}

// -------------------------------------------- self-loop + bias init pass ----
// out[i,:] = dis[i]^2 * hw[i,:] + bias[:]   (fully coalesced float4 stream)
__global__ __launch_bounds__(256) void gcn_agg_init(
        float* __restrict__ out, const float* __restrict__ hw,
        const float* __restrict__ dis, const float* __restrict__ bias,
        int n, int h) {
    int h4  = h >> 2;
    int idx = blockIdx.x * blockDim.x + threadIdx.x;
    if (idx >= n * h4) return;
    int row = idx / h4;
    int cj  = idx - row * h4;
    float sn = dis[row];
    sn = sn * sn;
    v4f v  = ((const v4f*)hw)[idx];
    v4f bv = ((const v4f*)bias)[cj];
    ((v4f*)out)[idx] = v * sn + bv;
}

// ------------------------------------------------ edge gather + scatter -----
// One wave per edge: lane loads float4 of hw[src], scales by dis[s]*dis[d],
// 4x global_atomic_add_f32 into agg[dst]. hw (25.6MB) lives in the 192MB L2,
// so the random gather is L2-bound, atomics RMW in L2.
__global__ __launch_bounds__(256) void gcn_scatter(
        float* __restrict__ agg, const float* __restrict__ hw,
        const float* __restrict__ dis, const int* __restrict__ src,
        const int* __restrict__ dst, int e, int h) {
    int wave = blockIdx.x * (blockDim.x >> 5) + (threadIdx.x >> 5);
    if (wave >= e) return;
    int lane = threadIdx.x & 31;
    int s = src[wave];
    int d = dst[wave];
    float nrm = dis[s] * dis[d];

    v4f m = ((const v4f*)(hw + (size_t)s * h))[lane];
    m *= nrm;
    float* drow = agg + (size_t)d * h + lane * 4;
    atomicAdd(drow + 0, m.x);
    atomicAdd(drow + 1, m.y);
    atomicAdd(drow + 2, m.z);
    atomicAdd(drow + 3, m.w);
}

// ----------------------------------------------------------------- driver ---
extern "C" void kernel_launch(void* const* d_in, const int* in_sizes, int n_in,
                              void* d_out, int out_size, void* d_ws, size_t ws_size,
                              hipStream_t stream) {
    const float* x  = (const float*)d_in[0];
    const int*   ei = (const int*)  d_in[1];   // [2,E]: row0=src, row1=dst
    const float* W1 = (const float*)d_in[2];
    const float* b1 = (const float*)d_in[3];
    const float* Wc = (const float*)d_in[4];   // [L,H,H]
    const float* bc = (const float*)d_in[5];   // [L,H]
    const float* W2 = (const float*)d_in[6];
    const float* b2 = (const float*)d_in[7];

    const int N = 50000, E = 800000, H = 128, L = 3;
    const int* srcv = ei;
    const int* dstv = ei + E;

    // carve workspace (deg, dis, two h ping-pong buffers, hw) ~77 MB
    char* ws = (char*)d_ws;
    size_t off = 0;
    auto carve = [&](size_t bytes) -> void* {
        void* p = ws + off;
        off = (off + bytes + 255) & ~(size_t)255;
        return p;
    };
    float* deg  = (float*)carve((size_t)N * 4);
    float* dis  = (float*)carve((size_t)N * 4);
    float* bufA = (float*)carve((size_t)N * H * 4);
    float* bufB = (float*)carve((size_t)N * H * 4);
    float* hw   = (float*)carve((size_t)N * H * 4);

    // degree / normalization
    gcn_deg_init <<<(N + 255) / 256, 256, 0, stream>>>(deg, N);
    gcn_deg_count<<<(E + 255) / 256, 256, 0, stream>>>(deg, dstv, E);
    gcn_rsqrt    <<<(N + 255) / 256, 256, 0, stream>>>(dis, deg, N);

    const int tilesM    = N / 16;              // 3125
    const int gemmBlks  = (tilesM + 7) / 8;    // 8 waves (row tiles) per block

    // h = relu(x @ W1 + b1)    [N,128]x[128,128]
    gcn_gemm_wmma<128, 8, true, true><<<gemmBlks, 256, 0, stream>>>(
        x, W1, b1, bufA, N);

    float* hcur = bufA;
    float* hnxt = bufB;
    for (int l = 0; l < L; ++l) {
        // hw = h @ Wc[l]       [N,128]x[128,128]
        gcn_gemm_wmma<128, 8, false, false><<<gemmBlks, 256, 0, stream>>>(
            hcur, Wc + (size_t)l * H * H, nullptr, hw, N);

        // h_next = self_norm * hw + bias
        int tot4 = N * (H / 4);
        gcn_agg_init<<<(tot4 + 255) / 256, 256, 0, stream>>>(
            hnxt, hw, dis, bc + (size_t)l * H, N, H);

        // h_next += scatter_add(norm * hw[src] -> dst)
        gcn_scatter<<<(E + 7) / 8, 256, 0, stream>>>(
            hnxt, hw, dis, srcv, dstv, E, H);

        float* t = hcur; hcur = hnxt; hnxt = t;
    }

    // out = h @ W2 + b2        [N,128]x[128,64]
    gcn_gemm_wmma<128, 4, false, true><<<gemmBlks, 256, 0, stream>>>(
        hcur, W2, b2, (float*)d_out, N);
}